// Ricker_3839700762671
// MI455X (gfx1250) — compile-verified
//
#include <hip/hip_runtime.h>
#include <stdint.h>

// Ricker stochastic scan, T = 4,194,304.
// Latency-bound serial recurrence: n' = n*exp(alpha*(1 - beta*n + f)) + e
// Rewritten as n' = fma(n, exp2(fma(nc, n, g_k)), e_k) with
//   g_k = K*(1 + bx*t + cx*t^2),  K = alpha*log2(e),  nc = -beta*K,  e_k = sigma*eps_k
// Single workgroup: lane 0 runs the dependent chain; all 256 lanes
// precompute g/e per chunk and drive the CDNA5 async LDS data mover
// (global_load_async_to_lds_b128 / global_store_async_from_lds_b32,
//  s_wait_asynccnt) for double-buffered streaming.

#define CHUNK 2048
#define NTHR  256

__device__ __forceinline__ uint32_t lds_off(const void* p) {
  // CDNA5 flat-aperture rule: low 32 bits of a flat LDS pointer == LDS byte offset.
  return (uint32_t)(uintptr_t)p;
}

__device__ __forceinline__ void wait_async_0() {
#if defined(__has_builtin) && __has_builtin(__builtin_amdgcn_s_wait_asynccnt)
  __builtin_amdgcn_s_wait_asynccnt(0);
#else
  asm volatile("s_wait_asynccnt 0x0" ::: "memory");
#endif
}

// Copy CHUNK floats (8 KB, 16B-aligned source) global -> LDS via async engine.
// 256 lanes x b128 x 2 packets. GVS addressing: saddr(64b SGPR) + vaddr(32b VGPR).
__device__ __forceinline__ void async_load_8kb(const float* gsrc, uint32_t lbase, int tid) {
  uint64_t s = (uint64_t)(uintptr_t)gsrc;
  uint32_t o0 = (uint32_t)tid * 16u;
  asm volatile("global_load_async_to_lds_b128 %0, %1, %2"
               :: "v"(lbase + o0), "v"(o0), "s"(s) : "memory");
  uint32_t o1 = o0 + (uint32_t)NTHR * 16u;
  asm volatile("global_load_async_to_lds_b128 %0, %1, %2"
               :: "v"(lbase + o1), "v"(o1), "s"(s) : "memory");
}

// Copy CHUNK floats LDS -> global. b32 packets: destination is out+1+c*CHUNK,
// which is only 4-byte aligned (out[0] = N0 shifts the trajectory by one float).
__device__ __forceinline__ void async_store_8kb(float* gdst, uint32_t lbase, int tid) {
  uint64_t s = (uint64_t)(uintptr_t)gdst;
#pragma unroll
  for (int p = 0; p < 8; ++p) {
    uint32_t o = (uint32_t)tid * 4u + (uint32_t)p * (NTHR * 4u);
    asm volatile("global_store_async_from_lds_b32 %0, %1, %2"
                 :: "v"(o), "v"(lbase + o), "s"(s) : "memory");
  }
}

__global__ __launch_bounds__(NTHR, 1)
void ricker_scan_kernel(const float* __restrict__ N0,
                        const float* __restrict__ Temp,
                        const float* __restrict__ sigma,
                        const float* __restrict__ eps,
                        const float* __restrict__ mp,
                        float* __restrict__ out, int T)
{
  __shared__ float rawT[2][CHUNK];   // raw Temp, transformed in place to g_k
  __shared__ float rawE[2][CHUNK];   // raw eps,  transformed in place to e_k
  __shared__ float outb[2][CHUNK];   // trajectory staging for async store-from-LDS

  const int tid = threadIdx.x;

  const float alpha = mp[0], beta = mp[1], bx = mp[2], cx = mp[3];
  const float K   = alpha * 1.44269504088896340736f;  // alpha * log2(e)
  const float gc1 = bx * K;
  const float gc2 = cx * K;
  const float nc  = -beta * K;
  const float sg  = sigma[0];
  float n = N0[0];
  if (tid == 0) out[0] = n;

  const int total = T - 1;
  const int FULLC = total / CHUNK;
  const int tail  = total - FULLC * CHUNK;

  const uint32_t lT[2] = { lds_off(&rawT[0][0]), lds_off(&rawT[1][0]) };
  const uint32_t lE[2] = { lds_off(&rawE[0][0]), lds_off(&rawE[1][0]) };
  const uint32_t lO[2] = { lds_off(&outb[0][0]), lds_off(&outb[1][0]) };

  // All 256 lanes: (t, eps) -> (g, sigma*eps) in place. Off the critical path.
  auto transform = [&](int bb) {
#pragma unroll 4
    for (int k = tid; k < CHUNK; k += NTHR) {
      float t  = rawT[bb][k];
      float ee = rawE[bb][k];
      rawT[bb][k] = __builtin_fmaf(t, __builtin_fmaf(gc2, t, gc1), K);
      rawE[bb][k] = sg * ee;
    }
  };

  // Lane 0 only: minimal dependent chain fma -> v_exp_f32 -> fma per step.
  // LDS loads/stores are independent of n and co-execute during TRANS stalls.
  auto serial = [&](int bb, int cnt) {
#pragma unroll 16
    for (int k = 0; k < cnt; ++k) {
      float a = __builtin_fmaf(nc, n, rawT[bb][k]);
      float p = __builtin_amdgcn_exp2f(a);
      n = __builtin_fmaf(n, p, rawE[bb][k]);
      outb[bb][k] = n;
    }
  };

  if (FULLC > 0) {
    // Prologue: land chunk 0, transform it, start prefetch of chunk 1.
    async_load_8kb(Temp, lT[0], tid);
    async_load_8kb(eps,  lE[0], tid);
    wait_async_0();
    __syncthreads();
    transform(0);
    if (FULLC > 1) {
      async_load_8kb(Temp + CHUNK, lT[1], tid);
      async_load_8kb(eps  + CHUNK, lE[1], tid);
    }
    __syncthreads();

    for (int c = 0; c < FULLC; ++c) {
      const int b = c & 1;
      if (tid == 0) serial(b, CHUNK);     // ~30K cycles: hides prefetch latency
      wait_async_0();                      // chunk c+1 landed; our stores drained
      __syncthreads();                     // outb[b] + raw[b^1] visible to all waves
      if (c + 1 < FULLC) {
        transform(b ^ 1);                  // prep next chunk's g/e (parallel)
        if (c + 2 < FULLC) {               // prefetch chunk c+2 into freed buffer
          async_load_8kb(Temp + (c + 2) * CHUNK, lT[b], tid);
          async_load_8kb(eps  + (c + 2) * CHUNK, lE[b], tid);
        }
      }
      async_store_8kb(out + 1 + c * CHUNK, lO[b], tid);  // drain chunk c results
      __syncthreads();                     // transform(b^1) visible before next serial
    }
  }

  // Tail chunk (also covers eps being one element shorter than Temp):
  // plain guarded loads/stores, no async OOB risk.
  wait_async_0();
  __syncthreads();
  if (tail > 0) {
    const int base = FULLC * CHUNK;
    for (int k = tid; k < tail; k += NTHR) {
      float t  = Temp[base + k];
      float ee = eps[base + k];
      rawT[0][k] = __builtin_fmaf(t, __builtin_fmaf(gc2, t, gc1), K);
      rawE[0][k] = sg * ee;
    }
    __syncthreads();
    if (tid == 0) serial(0, tail);
    __syncthreads();
    for (int k = tid; k < tail; k += NTHR) out[1 + base + k] = outb[0][k];
  }
}

extern "C" void kernel_launch(void* const* d_in, const int* in_sizes, int n_in,
                              void* d_out, int out_size, void* d_ws, size_t ws_size,
                              hipStream_t stream) {
  const float* N0    = (const float*)d_in[0];
  const float* Temp  = (const float*)d_in[1];
  const float* sigma = (const float*)d_in[2];
  const float* eps   = (const float*)d_in[3];
  const float* mp    = (const float*)d_in[4];
  float* out = (float*)d_out;
  const int T = in_sizes[1];
  (void)n_in; (void)out_size; (void)d_ws; (void)ws_size;
  ricker_scan_kernel<<<dim3(1), dim3(NTHR), 0, stream>>>(N0, Temp, sigma, eps, mp, out, T);
}